// PEAResNet_876173328851
// MI455X (gfx1250) — compile-verified
//
#include <hip/hip_runtime.h>
#include <hip/hip_bf16.h>

// ---------------- CDNA5 vector types ----------------
typedef __attribute__((ext_vector_type(2)))  float v2f;
typedef __attribute__((ext_vector_type(8)))  float v8f;
typedef __attribute__((ext_vector_type(4)))  unsigned int v4u;
typedef __attribute__((ext_vector_type(8)))  int v8i;
typedef __attribute__((ext_vector_type(4)))  int v4i;

// Problem constants (from reference)
#define BB   32
#define CC   256
#define HW   3136          // 56*56
#define GG   8
#define GSZ  32
#define NPIX 100352        // BB*HW
#define NTILE 1568         // NPIX/64, 64 | 3136 so tiles never cross batch rows
#define PLANE 802816       // CC*HW
#define GPLANE 100352      // GSZ*HW

// D = A(16x4) * B(4x16) + C, fp32 WMMA
__device__ __forceinline__ v8f wmma4(v2f a, v2f b, v8f c) {
  return __builtin_amdgcn_wmma_f32_16x16x4_f32(false, a, false, b, (short)0, c,
                                               false, false);
}

// ---------------- kernel 0: zero accumulators ----------------
__global__ void k_zero(float* p, int n) {
  int i = blockIdx.x * 256 + threadIdx.x;
  if (i < n) p[i] = 0.0f;
}

// ---------------- kernel 1: per-group sum_x / sum_xx via WMMA ----------------
// grid = 8 groups * 49 blocks, 256 threads (8 waves). Each wave: 4 tiles of 64 px.
__global__ void __launch_bounds__(256) k_stats(const float* __restrict__ feat,
                                               float* __restrict__ sumx,
                                               float* __restrict__ sumxx) {
  const int g    = blockIdx.x & 7;
  const int blk  = blockIdx.x >> 3;     // 0..48
  const int wave = threadIdx.x >> 5;
  const int lane = threadIdx.x & 31;
  const int li = lane & 15, hi = lane >> 4;

  v8f acc00 = {}, acc01 = {}, acc10 = {}, acc11 = {};
  float sx0 = 0.0f, sx1 = 0.0f;

  const int wv = blk * 8 + wave;        // 0..391 within group
  for (int t = wv; t < NTILE; t += 392) {
    const int pb = t * 64;
    const int b  = pb / HW;
    const int s  = pb - b * HW;
    const float* base = feat + (size_t)b * PLANE + (size_t)g * GPLANE + s;
    const float* p0 = base + (size_t)li * HW;          // channel li
    const float* p1 = base + (size_t)(li + 16) * HW;   // channel li+16
#pragma unroll
    for (int ks = 0; ks < 16; ++ks) {
      const int poff = ks * 4 + hi * 2;   // A/B layout: lanes 16-31 hold K+2
      v2f h0 = *(const v2f*)(p0 + poff);  // X[p..p+1][li]
      v2f h1 = *(const v2f*)(p1 + poff);  // X[p..p+1][li+16]
      sx0 += h0.x + h0.y;
      sx1 += h1.x + h1.y;
      acc00 = wmma4(h0, h0, acc00);       // i:0-15  x j:0-15
      acc01 = wmma4(h0, h1, acc01);       // i:0-15  x j:16-31
      acc10 = wmma4(h1, h0, acc10);       // i:16-31 x j:0-15
      acc11 = wmma4(h1, h1, acc11);       // i:16-31 x j:16-31
    }
  }

  // block-level LDS reduction, then few global atomics
  __shared__ float xr[1056];              // 1024 xx + 32 sx
  for (int e = threadIdx.x; e < 1056; e += 256) xr[e] = 0.0f;
  __syncthreads();
#pragma unroll
  for (int r = 0; r < 8; ++r) {
    const int i0 = r + 8 * hi;            // D layout: lanes 16-31 hold M+8
    atomicAdd(&xr[i0 * 32 + li],            acc00[r]);
    atomicAdd(&xr[i0 * 32 + li + 16],       acc01[r]);
    atomicAdd(&xr[(i0 + 16) * 32 + li],     acc10[r]);
    atomicAdd(&xr[(i0 + 16) * 32 + li + 16],acc11[r]);
  }
  atomicAdd(&xr[1024 + li],      sx0);
  atomicAdd(&xr[1024 + li + 16], sx1);
  __syncthreads();
  for (int e = threadIdx.x; e < 1024; e += 256)
    atomicAdd(&sumxx[g * 1024 + e], xr[e]);
  if (threadIdx.x < 32)
    atomicAdd(&sumx[g * 32 + threadIdx.x], xr[1024 + threadIdx.x]);
}

// ---------------- kernel 2: Newton-Schulz invsqrt/sqrt, fused M', bias ----------
__device__ __forceinline__ void mat_ns(float* A, float* Y, float* Z, float* T,
                                       float* U, float* red, float* scp, int tid) {
  // scale = 1/frobenius(A); spectrum of s*A lands in (0,1] -> NS converges
  float ss = 0.0f;
#pragma unroll
  for (int r = 0; r < 4; ++r) {
    int e = tid + 256 * r; int i = e >> 5, j = e & 31;
    float v = A[i * 33 + j]; ss += v * v;
  }
  red[tid] = ss; __syncthreads();
  for (int w = 128; w > 0; w >>= 1) {
    if (tid < w) red[tid] += red[tid + w];
    __syncthreads();
  }
  if (tid == 0) *scp = 1.0f / (sqrtf(red[0]) + 1e-12f);
  __syncthreads();
  const float s = *scp;
#pragma unroll
  for (int r = 0; r < 4; ++r) {
    int e = tid + 256 * r; int i = e >> 5, j = e & 31;
    Y[i * 33 + j] = A[i * 33 + j] * s;
    Z[i * 33 + j] = (i == j) ? 1.0f : 0.0f;
  }
  __syncthreads();
  for (int it = 0; it < 18; ++it) {
    // T = 1.5 I - 0.5 * Z @ Y
#pragma unroll
    for (int r = 0; r < 4; ++r) {
      int e = tid + 256 * r; int i = e >> 5, j = e & 31;
      float sum = 0.0f;
      for (int k = 0; k < 32; ++k) sum += Z[i * 33 + k] * Y[k * 33 + j];
      T[i * 33 + j] = ((i == j) ? 1.5f : 0.0f) - 0.5f * sum;
    }
    __syncthreads();
    // U = Y@T, A(reused as temp) = T@Z
#pragma unroll
    for (int r = 0; r < 4; ++r) {
      int e = tid + 256 * r; int i = e >> 5, j = e & 31;
      float su = 0.0f, sv = 0.0f;
      for (int k = 0; k < 32; ++k) {
        su += Y[i * 33 + k] * T[k * 33 + j];
        sv += T[i * 33 + k] * Z[k * 33 + j];
      }
      U[i * 33 + j] = su; A[i * 33 + j] = sv;
    }
    __syncthreads();
#pragma unroll
    for (int r = 0; r < 4; ++r) {
      int e = tid + 256 * r; int i = e >> 5, j = e & 31;
      Y[i * 33 + j] = U[i * 33 + j];
      Z[i * 33 + j] = A[i * 33 + j];
    }
    __syncthreads();
  }
  // Y ~ sqrt(s*A_in), Z ~ invsqrt(s*A_in)
}

__global__ void __launch_bounds__(256) k_transform(const float* __restrict__ sumx,
                                                   const float* __restrict__ sumxx,
                                                   const float* __restrict__ mu_src,
                                                   const float* __restrict__ css,
                                                   float* __restrict__ Mout,
                                                   float* __restrict__ biasout) {
  __shared__ float A[32 * 33], Y[32 * 33], Z[32 * 33], T[32 * 33], U[32 * 33];
  __shared__ float Winv[32 * 33];
  __shared__ float S[32 * 257];
  __shared__ float red[256];
  __shared__ float sc;
  __shared__ float mu_s[32];
  const int g = blockIdx.x, tid = threadIdx.x;

  if (tid < 32) mu_s[tid] = sumx[g * 32 + tid] * (1.0f / (float)NPIX);
  __syncthreads();
  // cov = (xx - N mu mu^T)/(N-1) + reg I
#pragma unroll
  for (int r = 0; r < 4; ++r) {
    int e = tid + 256 * r; int i = e >> 5, j = e & 31;
    float c = (sumxx[g * 1024 + e] - (float)NPIX * mu_s[i] * mu_s[j]) *
              (1.0f / (float)(NPIX - 1));
    if (i == j) c += 1e-5f;
    A[i * 33 + j] = c;
  }
  __syncthreads();
  mat_ns(A, Y, Z, T, U, red, &sc, tid);
  const float f_inv = sqrtf(sc);          // invsqrt(cov) = Z * sqrt(s)
#pragma unroll
  for (int r = 0; r < 4; ++r) {
    int e = tid + 256 * r; int i = e >> 5, j = e & 31;
    Winv[i * 33 + j] = Z[i * 33 + j] * f_inv;
  }
  __syncthreads();

  // source block: Bsrc = (cov_sqrt_src cov_sqrt_src^T) diagonal 32x32 block g
  for (int r = 0; r < 32; ++r)
    S[r * 257 + tid] = css[(size_t)(g * 32 + r) * 256 + tid];
  __syncthreads();
#pragma unroll
  for (int r = 0; r < 4; ++r) {
    int e = tid + 256 * r; int i = e >> 5, j = e & 31;
    float sum = 0.0f;
    for (int k = 0; k < 256; ++k) sum += S[i * 257 + k] * S[j * 257 + k];
    A[i * 33 + j] = sum;
  }
  __syncthreads();
  mat_ns(A, Y, Z, T, U, red, &sc, tid);
  const float f_src = 1.0f / sqrtf(sc);   // sqrt(Bsrc) = Y / sqrt(s)

  // M = Winv @ sqrt(Bsrc);  M' = 0.5(M + I);  bias = 0.5(mu_src - mu@M)
#pragma unroll
  for (int r = 0; r < 4; ++r) {
    int e = tid + 256 * r; int i = e >> 5, j = e & 31;
    float sum = 0.0f;
    for (int k = 0; k < 32; ++k) sum += Winv[i * 33 + k] * Y[k * 33 + j];
    float m = sum * f_src;
    T[i * 33 + j] = m;
    Mout[g * 1024 + i * 32 + j] = 0.5f * (m + ((i == j) ? 1.0f : 0.0f));
  }
  __syncthreads();
  if (tid < 32) {
    float s2 = 0.0f;
    for (int i = 0; i < 32; ++i) s2 += mu_s[i] * T[i * 33 + tid];
    biasout[g * 32 + tid] = 0.5f * (mu_src[g * 32 + tid] - s2);
  }
}

// ---------------- kernel 3: out = X @ M' + bias via WMMA + TDM -----------------
// TDM load of M' (32x32 f32) into LDS with pitch-33 padding (pad 1 dword / 32).
__device__ __forceinline__ void tdm_load_M(const float* Mg, unsigned ldsOff) {
  unsigned long long ga = (unsigned long long)(uintptr_t)Mg;
  v4u g0;
  g0.x = 1u;                                   // count=1, user descriptor
  g0.y = ldsOff;                               // lds_addr (bytes)
  g0.z = (unsigned)ga;                         // global_addr[31:0]
  g0.w = (unsigned)((ga >> 32) & 0x1FFFFFFu) | (2u << 30);  // addr[56:32] | type=2
  v8i g1 = {
      0x01120000,       // data_size=4B, pad_enable, pad_interval=32dw, pad=1dw
      (int)(32u << 16), // tensor_dim0 = 32
      (int)(32u << 16), // tensor_dim1 = 32
      (int)(32u << 16), // tile_dim0 = 32
      32,               // tile_dim1 = 32, tile_dim2 = 0
      32,               // tensor_dim0_stride = 32
      0, 0};            // tensor_dim1_stride = 0 (2D)
  v4i zz4 = {0, 0, 0, 0};
  v8i zz8 = {0, 0, 0, 0, 0, 0, 0, 0};
  __builtin_amdgcn_tensor_load_to_lds(g0, g1, zz4, zz4, zz8, 0);
}

__global__ void __launch_bounds__(256) k_apply(const float* __restrict__ feat,
                                               const float* __restrict__ Mw,
                                               const float* __restrict__ biasw,
                                               float* __restrict__ out) {
  __shared__ float Mlds[32 * 33];
  __shared__ float bias_l[32];
  __shared__ float stage[4 * 32 * 66];    // one 64px x 32ch tile per wave-pair

  const int g    = blockIdx.x & 7;
  const int blk  = blockIdx.x >> 3;       // 0..48
  const int wave = threadIdx.x >> 5;
  const int lane = threadIdx.x & 31;
  const int li = lane & 15, hi = lane >> 4;
  const int pi = wave >> 1;               // pair id 0..3
  const int sp = wave & 1;                // half within pair

  if (threadIdx.x < 32) {
    unsigned ldsOff = (unsigned)(unsigned long long)(uintptr_t)(void*)Mlds;
    tdm_load_M(Mw + g * 1024, ldsOff);    // async tensor DMA -> LDS
    bias_l[lane] = biasw[g * 32 + lane];
    __builtin_amdgcn_s_wait_tensorcnt(0);
  }
  __syncthreads();

  // hoist loop-invariant B fragments of M' (B 4x16: vgpr q -> K=q / K=q+2)
  v2f bf[2][8];
#pragma unroll
  for (int jh = 0; jh < 2; ++jh)
#pragma unroll
    for (int kc = 0; kc < 8; ++kc) {
      const int k0 = kc * 4 + 2 * hi;
      const int j  = li + 16 * jh;
      v2f t; t.x = Mlds[k0 * 33 + j]; t.y = Mlds[(k0 + 1) * 33 + j];
      bf[jh][kc] = t;
    }
  const float b0 = bias_l[li], b1 = bias_l[li + 16];
  float* stg = stage + pi * (32 * 66);

  const int pw = blk * 4 + pi;            // pair index 0..195 within group
  for (int t = pw; t < NTILE; t += 196) { // exactly 8 tiles per pair (uniform)
    const int pb = t * 64;
    const int b  = pb / HW;
    const int s  = pb - b * HW;
    const float* inb = feat + (size_t)b * PLANE + (size_t)g * GPLANE + s;
    float*      outb = out  + (size_t)b * PLANE + (size_t)g * GPLANE + s;

#pragma unroll
    for (int su = 0; su < 2; ++su) {
      const int sub = sp * 2 + su;        // 16-pixel subtile 0..3
      const float* colb = inb + sub * 16 + li;   // pixel = lane row of A
      v8f a0 = {b0, b0, b0, b0, b0, b0, b0, b0};
      v8f a1 = {b1, b1, b1, b1, b1, b1, b1, b1};
#pragma unroll
      for (int kc = 0; kc < 8; ++kc) {
        const int c0 = kc * 4 + 2 * hi;   // A 16x4: vgpr q -> K=q / K=q+2
        v2f a;
        a.x = colb[(size_t)c0 * HW];
        a.y = colb[(size_t)(c0 + 1) * HW];
        a0 = wmma4(a, bf[0][kc], a0);
        a1 = wmma4(a, bf[1][kc], a1);
      }
      // D(m=pixel r+8*hi, n=channel li+16*jh) -> LDS [channel][pixel]
      const int px0 = sub * 16 + 8 * hi;
      float* d0 = stg + li * 66 + px0;
      float* d1 = stg + (li + 16) * 66 + px0;
#pragma unroll
      for (int r = 0; r < 8; r += 2) {
        *(v2f*)(d0 + r) = (v2f){a0[r], a0[r + 1]};
        *(v2f*)(d1 + r) = (v2f){a1[r], a1[r + 1]};
      }
    }
    __syncthreads();
    // coalesced write-back: each wave of the pair stores 16 channel rows
#pragma unroll
    for (int cc = 0; cc < 16; ++cc) {
      const int c = sp * 16 + cc;
      v2f v = *(const v2f*)(stg + c * 66 + 2 * lane);
      *(v2f*)(outb + (size_t)c * HW + 2 * lane) = v;
    }
    __syncthreads();
  }
}

// ---------------- launch ----------------
extern "C" void kernel_launch(void* const* d_in, const int* in_sizes, int n_in,
                              void* d_out, int out_size, void* d_ws, size_t ws_size,
                              hipStream_t stream) {
  const float* feat   = (const float*)d_in[0];
  const float* mu_src = (const float*)d_in[1];
  const float* css    = (const float*)d_in[2];
  float* out = (float*)d_out;

  float* wsf   = (float*)d_ws;
  float* sumx  = wsf;            // 256
  float* sumxx = wsf + 256;      // 8192
  float* Mp    = wsf + 8448;     // 8192  (M' = 0.5(I+M))
  float* biasw = wsf + 16640;    // 256   (0.5(mu_src - mu@M))

  k_zero<<<(8448 + 255) / 256, 256, 0, stream>>>(wsf, 8448);
  k_stats<<<8 * 49, 256, 0, stream>>>(feat, sumx, sumxx);
  k_transform<<<8, 256, 0, stream>>>(sumx, sumxx, mu_src, css, Mp, biasw);
  k_apply<<<8 * 49, 256, 0, stream>>>(feat, Mp, biasw, out);
}